// Decoder_85057532330299
// MI455X (gfx1250) — compile-verified
//
#include <hip/hip_runtime.h>
#include <math.h>

typedef float v2f __attribute__((ext_vector_type(2)));
typedef float v8f __attribute__((ext_vector_type(8)));

#define HID   1024
#define VOC   32000
#define NSTEP 128

// D = A(16x4) * B(4x16) + C, all f32, wave32.
// 8 args: (neg_a, A, neg_b, B, c_mod, C, reuse_a, reuse_b)
__device__ __forceinline__ v8f wmma4(v2f a, v2f b, v8f c) {
  return __builtin_amdgcn_wmma_f32_16x16x4_f32(false, a, false, b, (short)0, c,
                                               false, false);
}

__device__ __forceinline__ float sigm(float x) {
  return 1.0f / (1.0f + __expf(-x));
}

// ---------------------------------------------------------------------------
// h_init = W_up @ (cv0*cv1) + b_up ; h0=c0=h1=c1=h_init ; tok = y[0]
// 64 blocks x 256 threads (8 waves, k-split of 128 each)
// ---------------------------------------------------------------------------
__global__ void k_init(const float* __restrict__ cv,
                       const float* __restrict__ Wup,
                       const float* __restrict__ bup,
                       const int*   __restrict__ y,
                       float* __restrict__ h0, float* __restrict__ c0,
                       float* __restrict__ h1, float* __restrict__ c1,
                       float* __restrict__ tok) {
  __shared__ float ctx[HID];
  __shared__ float red[8 * 32 * 8];
  const int tid = threadIdx.x;
  for (int i = tid; i < HID; i += 256) ctx[i] = cv[i] * cv[HID + i];
  __syncthreads();

  const int w = tid >> 5, l = tid & 31;
  const int hi = l >> 4, m = l & 15;
  const int rb = blockIdx.x * 16;
  const float* wp = Wup + (size_t)(rb + m) * HID + w * 128 + 2 * hi;
  const float* vp = ctx + w * 128 + 2 * hi;
  v8f acc = {};
#pragma unroll 4
  for (int kk = 0; kk < 128; kk += 4)
    acc = wmma4(*(const v2f*)(wp + kk), *(const v2f*)(vp + kk), acc);

#pragma unroll
  for (int r = 0; r < 8; ++r) red[(w * 32 + l) * 8 + r] = acc[r];
  __syncthreads();

  if (tid < 32) {
    float s[8];
#pragma unroll
    for (int r = 0; r < 8; ++r) {
      float t = 0.f;
      for (int ww = 0; ww < 8; ++ww) t += red[(ww * 32 + l) * 8 + r];
      s[r] = t;
    }
    if (l == 0 || l == 16) {
      const int base = rb + (l == 16 ? 8 : 0);
      for (int r = 0; r < 8; ++r) {
        float v = s[r] + bup[base + r];
        h0[base + r] = v; c0[base + r] = v;
        h1[base + r] = v; c1[base + r] = v;
      }
    }
    if (blockIdx.x == 0 && tid == 0) *tok = (float)y[0];
  }
}

// ---------------------------------------------------------------------------
// LSTM layer 0 (input = scalar token). 64 blocks x 256 threads.
// Each block: 16 output j's; 8 waves k-split; 4 gate accumulators per wave.
// ---------------------------------------------------------------------------
__global__ void k_cell0(const float* __restrict__ Whh,
                        const float* __restrict__ Wih,
                        const float* __restrict__ bih,
                        const float* __restrict__ bhh,
                        const float* __restrict__ h_in,
                        const float* __restrict__ c_in,
                        float* __restrict__ h_out,
                        float* __restrict__ c_out,
                        const float* __restrict__ tok) {
  __shared__ float hs[HID];
  __shared__ float red[8 * 4 * 32 * 8];  // 32 KB
  const int tid = threadIdx.x;
  for (int i = tid; i < HID; i += 256) hs[i] = h_in[i];
  __syncthreads();

  const int w = tid >> 5, l = tid & 31;
  const int hi = l >> 4, m = l & 15;
  const int jb = blockIdx.x * 16;
  const int kb = w * 128 + 2 * hi;
  const float* w0 = Whh + (size_t)(jb + m) * HID + kb;
  const float* w1 = w0 + (size_t)1024 * HID;
  const float* w2 = w0 + (size_t)2048 * HID;
  const float* w3 = w0 + (size_t)3072 * HID;
  const float* vp = hs + kb;
  v8f ai = {}, af = {}, ag = {}, ao = {};
#pragma unroll 2
  for (int kk = 0; kk < 128; kk += 4) {
    v2f b = *(const v2f*)(vp + kk);
    ai = wmma4(*(const v2f*)(w0 + kk), b, ai);
    af = wmma4(*(const v2f*)(w1 + kk), b, af);
    ag = wmma4(*(const v2f*)(w2 + kk), b, ag);
    ao = wmma4(*(const v2f*)(w3 + kk), b, ao);
  }
#pragma unroll
  for (int r = 0; r < 8; ++r) {
    red[((w * 4 + 0) * 32 + l) * 8 + r] = ai[r];
    red[((w * 4 + 1) * 32 + l) * 8 + r] = af[r];
    red[((w * 4 + 2) * 32 + l) * 8 + r] = ag[r];
    red[((w * 4 + 3) * 32 + l) * 8 + r] = ao[r];
  }
  __syncthreads();

  if (tid < 32) {
    float si[8], sf[8], sg[8], so[8];
#pragma unroll
    for (int r = 0; r < 8; ++r) {
      float a = 0.f, b = 0.f, c = 0.f, d = 0.f;
      for (int ww = 0; ww < 8; ++ww) {
        a += red[((ww * 4 + 0) * 32 + l) * 8 + r];
        b += red[((ww * 4 + 1) * 32 + l) * 8 + r];
        c += red[((ww * 4 + 2) * 32 + l) * 8 + r];
        d += red[((ww * 4 + 3) * 32 + l) * 8 + r];
      }
      si[r] = a; sf[r] = b; sg[r] = c; so[r] = d;
    }
    if (l == 0 || l == 16) {
      const float tk = *tok;
      const int base = jb + (l == 16 ? 8 : 0);
      for (int r = 0; r < 8; ++r) {
        const int j = base + r;
        float gi = si[r] + Wih[j]        * tk + bih[j]        + bhh[j];
        float gf = sf[r] + Wih[1024 + j] * tk + bih[1024 + j] + bhh[1024 + j];
        float gg = sg[r] + Wih[2048 + j] * tk + bih[2048 + j] + bhh[2048 + j];
        float go = so[r] + Wih[3072 + j] * tk + bih[3072 + j] + bhh[3072 + j];
        float I = sigm(gi), F = sigm(gf), G = tanhf(gg), O = sigm(go);
        float cn = F * c_in[j] + I * G;
        c_out[j] = cn;
        h_out[j] = O * tanhf(cn);
      }
    }
  }
}

// ---------------------------------------------------------------------------
// LSTM layer 1 (input = h0_new). Both matvecs accumulate into one C tile.
// ---------------------------------------------------------------------------
__global__ void k_cell1(const float* __restrict__ Wih,
                        const float* __restrict__ Whh,
                        const float* __restrict__ bih,
                        const float* __restrict__ bhh,
                        const float* __restrict__ x_in,
                        const float* __restrict__ h_in,
                        const float* __restrict__ c_in,
                        float* __restrict__ h_out,
                        float* __restrict__ c_out) {
  __shared__ float xs[HID];
  __shared__ float hs[HID];
  __shared__ float red[8 * 4 * 32 * 8];
  const int tid = threadIdx.x;
  for (int i = tid; i < HID; i += 256) { xs[i] = x_in[i]; hs[i] = h_in[i]; }
  __syncthreads();

  const int w = tid >> 5, l = tid & 31;
  const int hi = l >> 4, m = l & 15;
  const int jb = blockIdx.x * 16;
  const int kb = w * 128 + 2 * hi;
  const float* wi0 = Wih + (size_t)(jb + m) * HID + kb;
  const float* wi1 = wi0 + (size_t)1024 * HID;
  const float* wi2 = wi0 + (size_t)2048 * HID;
  const float* wi3 = wi0 + (size_t)3072 * HID;
  const float* wh0 = Whh + (size_t)(jb + m) * HID + kb;
  const float* wh1 = wh0 + (size_t)1024 * HID;
  const float* wh2 = wh0 + (size_t)2048 * HID;
  const float* wh3 = wh0 + (size_t)3072 * HID;
  const float* xp = xs + kb;
  const float* hp = hs + kb;
  v8f ai = {}, af = {}, ag = {}, ao = {};
#pragma unroll 2
  for (int kk = 0; kk < 128; kk += 4) {
    v2f bx = *(const v2f*)(xp + kk);
    v2f bh = *(const v2f*)(hp + kk);
    ai = wmma4(*(const v2f*)(wi0 + kk), bx, ai);
    ai = wmma4(*(const v2f*)(wh0 + kk), bh, ai);
    af = wmma4(*(const v2f*)(wi1 + kk), bx, af);
    af = wmma4(*(const v2f*)(wh1 + kk), bh, af);
    ag = wmma4(*(const v2f*)(wi2 + kk), bx, ag);
    ag = wmma4(*(const v2f*)(wh2 + kk), bh, ag);
    ao = wmma4(*(const v2f*)(wi3 + kk), bx, ao);
    ao = wmma4(*(const v2f*)(wh3 + kk), bh, ao);
  }
#pragma unroll
  for (int r = 0; r < 8; ++r) {
    red[((w * 4 + 0) * 32 + l) * 8 + r] = ai[r];
    red[((w * 4 + 1) * 32 + l) * 8 + r] = af[r];
    red[((w * 4 + 2) * 32 + l) * 8 + r] = ag[r];
    red[((w * 4 + 3) * 32 + l) * 8 + r] = ao[r];
  }
  __syncthreads();

  if (tid < 32) {
    float si[8], sf[8], sg[8], so[8];
#pragma unroll
    for (int r = 0; r < 8; ++r) {
      float a = 0.f, b = 0.f, c = 0.f, d = 0.f;
      for (int ww = 0; ww < 8; ++ww) {
        a += red[((ww * 4 + 0) * 32 + l) * 8 + r];
        b += red[((ww * 4 + 1) * 32 + l) * 8 + r];
        c += red[((ww * 4 + 2) * 32 + l) * 8 + r];
        d += red[((ww * 4 + 3) * 32 + l) * 8 + r];
      }
      si[r] = a; sf[r] = b; sg[r] = c; so[r] = d;
    }
    if (l == 0 || l == 16) {
      const int base = jb + (l == 16 ? 8 : 0);
      for (int r = 0; r < 8; ++r) {
        const int j = base + r;
        float gi = si[r] + bih[j]        + bhh[j];
        float gf = sf[r] + bih[1024 + j] + bhh[1024 + j];
        float gg = sg[r] + bih[2048 + j] + bhh[2048 + j];
        float go = so[r] + bih[3072 + j] + bhh[3072 + j];
        float I = sigm(gi), F = sigm(gf), G = tanhf(gg), O = sigm(go);
        float cn = F * c_in[j] + I * G;
        c_out[j] = cn;
        h_out[j] = O * tanhf(cn);
      }
    }
  }
}

// ---------------------------------------------------------------------------
// preds = relu(W_out @ h1 + b_out). 250 blocks x 256 thr; 1 row-group / wave.
// ---------------------------------------------------------------------------
__global__ void k_out(const float* __restrict__ W,
                      const float* __restrict__ b,
                      const float* __restrict__ h,
                      float* __restrict__ out) {
  __shared__ float hs[HID];
  const int tid = threadIdx.x;
  for (int i = tid; i < HID; i += 256) hs[i] = h[i];
  __syncthreads();

  const int w = tid >> 5, l = tid & 31;
  const int hi = l >> 4, m = l & 15;
  const int rb = (blockIdx.x * 8 + w) * 16;  // 2000 row-groups = 32000 rows
  const float* wp = W + (size_t)(rb + m) * HID + 2 * hi;
  const float* vp = hs + 2 * hi;
  v8f acc = {};
#pragma unroll 4
  for (int k = 0; k < HID; k += 4)
    acc = wmma4(*(const v2f*)(wp + k), *(const v2f*)(vp + k), acc);

  if (l == 0 || l == 16) {
    const int base = rb + (l == 16 ? 8 : 0);
    for (int r = 0; r < 8; ++r) {
      float v = acc[r] + b[base + r];
      out[base + r] = v > 0.f ? v : 0.f;
    }
  }
}

// ---------------------------------------------------------------------------
// argmax (next token) + logsumexp for this step. 1 block x 256 threads.
// ---------------------------------------------------------------------------
__global__ void k_argmax(const float* __restrict__ logits,
                         float* __restrict__ tok,
                         float* __restrict__ lse) {
  __shared__ float smax[256];
  __shared__ int   sidx[256];
  __shared__ float ssum[256];
  const int tid = threadIdx.x;
  float mv = -1e30f; int mi = 0;
  for (int i = tid; i < VOC; i += 256) {
    float v = logits[i];
    if (v > mv) { mv = v; mi = i; }
  }
  smax[tid] = mv; sidx[tid] = mi;
  __syncthreads();
  for (int s = 128; s > 0; s >>= 1) {
    if (tid < s) {
      float a = smax[tid], bv = smax[tid + s];
      if (bv > a || (bv == a && sidx[tid + s] < sidx[tid])) {
        smax[tid] = bv; sidx[tid] = sidx[tid + s];
      }
    }
    __syncthreads();
  }
  const float M = smax[0];
  float s = 0.f;
  for (int i = tid; i < VOC; i += 256) s += __expf(logits[i] - M);
  ssum[tid] = s;
  __syncthreads();
  for (int st = 128; st > 0; st >>= 1) {
    if (tid < st) ssum[tid] += ssum[tid + st];
    __syncthreads();
  }
  if (tid == 0) {
    *tok = (float)sidx[0];
    *lse = M + __logf(ssum[0]);
  }
}

// ---------------------------------------------------------------------------
// log_softmax finalize: out[t][v] -= lse[t]
// ---------------------------------------------------------------------------
__global__ void k_norm(float* __restrict__ out, const float* __restrict__ lse) {
  const int i = blockIdx.x * 256 + threadIdx.x;
  const int total = NSTEP * VOC;
  if (i < total) out[i] -= lse[i / VOC];
}

// ---------------------------------------------------------------------------
extern "C" void kernel_launch(void* const* d_in, const int* in_sizes, int n_in,
                              void* d_out, int out_size, void* d_ws,
                              size_t ws_size, hipStream_t stream) {
  (void)in_sizes; (void)n_in; (void)out_size; (void)ws_size;
  const int*   y    = (const int*)d_in[0];
  const float* cv   = (const float*)d_in[1];
  // d_in[2] = stride (fixed 128)
  const float* Wup  = (const float*)d_in[3];
  const float* bup  = (const float*)d_in[4];
  const float* Wih0 = (const float*)d_in[5];
  const float* Whh0 = (const float*)d_in[6];
  const float* bih0 = (const float*)d_in[7];
  const float* bhh0 = (const float*)d_in[8];
  const float* Wih1 = (const float*)d_in[9];
  const float* Whh1 = (const float*)d_in[10];
  const float* bih1 = (const float*)d_in[11];
  const float* bhh1 = (const float*)d_in[12];
  const float* Wout = (const float*)d_in[13];
  const float* bout = (const float*)d_in[14];
  float* out = (float*)d_out;
  float* ws  = (float*)d_ws;

  // workspace layout (floats): ping-pong state buffers + token + per-step lse
  float* h0b = ws;               // [2][1024]
  float* c0b = ws + 2048;        // [2][1024]
  float* h1b = ws + 4096;        // [2][1024]
  float* c1b = ws + 6144;        // [2][1024]
  float* tok = ws + 8192;        // [1]
  float* lse = ws + 8256;        // [128]

  k_init<<<64, 256, 0, stream>>>(cv, Wup, bup, y, h0b, c0b, h1b, c1b, tok);

  for (int t = 0; t < NSTEP; ++t) {
    const int p = t & 1, q = p ^ 1;
    k_cell0<<<64, 256, 0, stream>>>(Whh0, Wih0, bih0, bhh0,
                                    h0b + p * HID, c0b + p * HID,
                                    h0b + q * HID, c0b + q * HID, tok);
    k_cell1<<<64, 256, 0, stream>>>(Wih1, Whh1, bih1, bhh1,
                                    h0b + q * HID,
                                    h1b + p * HID, c1b + p * HID,
                                    h1b + q * HID, c1b + q * HID);
    k_out<<<250, 256, 0, stream>>>(Wout, bout, h1b + q * HID,
                                   out + (size_t)t * VOC);
    k_argmax<<<1, 256, 0, stream>>>(out + (size_t)t * VOC, tok, lse + t);
  }
  k_norm<<<(NSTEP * VOC + 255) / 256, 256, 0, stream>>>(out, lse);
}